// MoE_49435073577414
// MI455X (gfx1250) — compile-verified
//
#include <hip/hip_runtime.h>

#define NUM_EXPERTS 8
#define HID 1024
#define INTER 4096
#define NTOK 2048

typedef __attribute__((ext_vector_type(2))) float v2f;
typedef __attribute__((ext_vector_type(8))) float v8f;
typedef __attribute__((ext_vector_type(4))) unsigned v4u;
typedef __attribute__((ext_vector_type(4))) int v4i;
typedef __attribute__((ext_vector_type(8))) int v8i;

typedef __attribute__((address_space(3))) void lds_void;

static __device__ __forceinline__ unsigned lds_addr_of(void* p) {
  return (unsigned)(unsigned long long)(lds_void*)p;  // LDS byte offset
}

static __device__ __forceinline__ v8f wmma_f32(v2f a, v2f b, v8f c) {
  // 8 args: (neg_a, A, neg_b, B, c_mod, C, reuse_a, reuse_b)
  return __builtin_amdgcn_wmma_f32_16x16x4_f32(false, a, false, b, (short)0, c,
                                               false, false);
}

// ---------------------------------------------------------------------------
// TDM: DMA one 64-row x 32-col f32 tile (global row stride = stride_elems)
// into LDS at byte offset lds_off. pad_enable inserts 4 dwords after every 32
// dwords stored -> LDS row stride of 36 floats (even => b64-aligned frags,
// 36*r mod 64 distinct over 16 rows => conflict-free fragment reads).
// ---------------------------------------------------------------------------
static __device__ __forceinline__ void tdm_load_tile_64x32(unsigned lds_off,
                                                           const float* gsrc,
                                                           unsigned stride_elems) {
  const unsigned long long ga = (unsigned long long)gsrc;
  v4u g0;
  g0[0] = 1u;                                 // count=1 (user descriptor)
  g0[1] = lds_off;                            // lds_addr (bytes)
  g0[2] = (unsigned)(ga & 0xffffffffull);     // global_addr[31:0]
  g0[3] = (unsigned)(ga >> 32) | (2u << 30);  // global_addr[56:32] | type=2
  v8i g1;
  // data_size=2 (4B) | pad_enable | pad_interval=4 (32 dw) | pad_amount=3 (4 dw)
  g1[0] = (2 << 16) | (1 << 20) | (4 << 22) | (3 << 25);
  g1[1] = 0;                                  // abar=0, tensor_dim0[15:0] (dim=1<<30)
  g1[2] = 0x4000;                             // tensor_dim0[31:16]
  g1[3] = 0x4000 | (32 << 16);                // tensor_dim1[31:16] | tile_dim0=32
  g1[4] = 64;                                 // tile_dim1=64, tile_dim2=0
  g1[5] = (int)stride_elems;                  // tensor_dim0_stride[31:0]
  g1[6] = 0;
  g1[7] = 0;
  v4i z4 = {0, 0, 0, 0};
#if __clang_major__ >= 23
  v8i z8 = {0, 0, 0, 0, 0, 0, 0, 0};
  __builtin_amdgcn_tensor_load_to_lds(g0, g1, z4, z4, z8, 0);
#else
  __builtin_amdgcn_tensor_load_to_lds(g0, g1, z4, z4, 0);
#endif
}

// ---------------------------------------------------------------------------
// Router: logits = x . gate_w ; renormalized top-2 softmax -> coef[T][8]
// ---------------------------------------------------------------------------
__global__ __launch_bounds__(256) void moe_router_k(const float* __restrict__ X,
                                                    const float* __restrict__ GW,
                                                    float* __restrict__ coef) {
  __shared__ float logits[NUM_EXPERTS];
  const int t = blockIdx.x;
  const int tid = threadIdx.x;
  if (tid < NUM_EXPERTS) logits[tid] = 0.0f;
  __syncthreads();

  float p[NUM_EXPERTS];
#pragma unroll
  for (int e = 0; e < NUM_EXPERTS; ++e) p[e] = 0.0f;

  for (int k = tid; k < HID; k += 256) {
    const float x = X[(size_t)t * HID + k];
#pragma unroll
    for (int e = 0; e < NUM_EXPERTS; ++e) p[e] += x * GW[(size_t)k * NUM_EXPERTS + e];
  }
#pragma unroll
  for (int e = 0; e < NUM_EXPERTS; ++e) atomicAdd(&logits[e], p[e]);
  __syncthreads();

  if (tid == 0) {
    int ia = -1, ib = -1;
    float la = -3.4e38f, lb = -3.4e38f;
#pragma unroll
    for (int e = 0; e < NUM_EXPERTS; ++e) {
      const float l = logits[e];
      if (l > la) { ib = ia; lb = la; ia = e; la = l; }
      else if (l > lb) { ib = e; lb = l; }
    }
    const float eb = __expf(lb - la);
    const float inv = 1.0f / (1.0f + eb);
#pragma unroll
    for (int e = 0; e < NUM_EXPERTS; ++e) {
      float c = 0.0f;
      if (e == ia) c = inv;
      if (e == ib) c = eb * inv;
      coef[(size_t)t * NUM_EXPERTS + e] = c;
    }
  }
}

__global__ __launch_bounds__(256) void moe_zero_k(float4* __restrict__ p, int n4) {
  const int i = blockIdx.x * 256 + threadIdx.x;
  if (i < n4) p[i] = make_float4(0.f, 0.f, 0.f, 0.f);
}

// ---------------------------------------------------------------------------
// GEMM1+3 fused: act[T,INTER] = silu(X @ W1) * (X @ W3) for one expert.
// Block tile 64x128, 8 waves; wave = 16x64 strip for both matrices.
// A tile: TDM -> LDS (stride 36). B tiles: reg-pipelined global loads, stored
// K-transposed in LDS (bT[n][k], stride 34) so each fragment is one b64 load.
// ---------------------------------------------------------------------------
__global__ __launch_bounds__(256) void moe_gemm13_k(const float* __restrict__ X,
                                                    const float* __restrict__ W1,
                                                    const float* __restrict__ W3,
                                                    float* __restrict__ act) {
  constexpr int BK = 32;
  __shared__ float aT[64 * 36];
  __shared__ float b1T[128 * 34];
  __shared__ float b3T[128 * 34];

  const int tid = threadIdx.x;
  const int lane = tid & 31;
  const int wave = tid >> 5;
  const int m0 = blockIdx.y * 64;
  const int n0 = blockIdx.x * 128;
  const int wm = (wave & 3) * 16;
  const int wn = (wave >> 2) * 64;
  const int lm = lane & 15;
  const int lg = lane >> 4;

  const int aBase = (wm + lm) * 36 + 2 * lg;
  const int bBase = (wn + lm) * 34 + 2 * lg;
  const unsigned aOff = lds_addr_of(aT);

  v8f acc1[4] = {};
  v8f acc3[4] = {};
  float4 rb1[4], rb3[4];

  // Preload chunk 0 of B into registers.
#pragma unroll
  for (int i = 0; i < 4; ++i) {
    const int f = tid * 4 + i * 1024;
    const int r = f >> 7, c = f & 127;
    const size_t g = (size_t)r * INTER + n0 + c;
    rb1[i] = *(const float4*)(W1 + g);
    rb3[i] = *(const float4*)(W3 + g);
  }

  for (int kk = 0; kk < HID; kk += BK) {
    // A tile via Tensor Data Mover (one wave issues the DMA).
    if (wave == 0) tdm_load_tile_64x32(aOff, X + (size_t)m0 * HID + kk, HID);

    // Store current B registers K-transposed into LDS.
#pragma unroll
    for (int i = 0; i < 4; ++i) {
      const int f = tid * 4 + i * 1024;
      const int r = f >> 7, c = f & 127;
      b1T[(c + 0) * 34 + r] = rb1[i].x;
      b1T[(c + 1) * 34 + r] = rb1[i].y;
      b1T[(c + 2) * 34 + r] = rb1[i].z;
      b1T[(c + 3) * 34 + r] = rb1[i].w;
      b3T[(c + 0) * 34 + r] = rb3[i].x;
      b3T[(c + 1) * 34 + r] = rb3[i].y;
      b3T[(c + 2) * 34 + r] = rb3[i].z;
      b3T[(c + 3) * 34 + r] = rb3[i].w;
    }
    if (wave == 0) __builtin_amdgcn_s_wait_tensorcnt(0);
    __syncthreads();

    // Prefetch next chunk of B into registers (overlaps the WMMA loop).
    if (kk + BK < HID) {
#pragma unroll
      for (int i = 0; i < 4; ++i) {
        const int f = tid * 4 + i * 1024;
        const int r = f >> 7, c = f & 127;
        const size_t g = (size_t)(kk + BK + r) * INTER + n0 + c;
        rb1[i] = *(const float4*)(W1 + g);
        rb3[i] = *(const float4*)(W3 + g);
      }
    }

#pragma unroll
    for (int k4 = 0; k4 < BK; k4 += 4) {
      const v2f a = *(const v2f*)&aT[aBase + k4];
#pragma unroll
      for (int nt = 0; nt < 4; ++nt) {
        const int off = bBase + nt * 16 * 34 + k4;
        const v2f b1 = *(const v2f*)&b1T[off];
        acc1[nt] = wmma_f32(a, b1, acc1[nt]);
        const v2f b3 = *(const v2f*)&b3T[off];
        acc3[nt] = wmma_f32(a, b3, acc3[nt]);
      }
    }
    __syncthreads();
  }

  // Epilogue: act = silu(h1) * h3
#pragma unroll
  for (int nt = 0; nt < 4; ++nt) {
    const int n = n0 + wn + nt * 16 + lm;
#pragma unroll
    for (int v = 0; v < 8; ++v) {
      const int m = m0 + wm + v + 8 * lg;
      const float h1 = acc1[nt][v];
      const float h3 = acc3[nt][v];
      const float s = h1 / (1.0f + __expf(-h1));
      act[(size_t)m * INTER + n] = s * h3;
    }
  }
}

// ---------------------------------------------------------------------------
// GEMM2: out[T,HID] += coef[:,e] * (act @ W2). Split-K over gridDim.z with
// global fp32 atomics (out is zeroed first).
// ---------------------------------------------------------------------------
__global__ __launch_bounds__(256) void moe_gemm2_k(const float* __restrict__ A,
                                                   const float* __restrict__ W2,
                                                   const float* __restrict__ coef,
                                                   int e,
                                                   float* __restrict__ out) {
  constexpr int BK = 32, KSPLIT = 4;
  __shared__ float aT[64 * 36];
  __shared__ float bT[128 * 34];

  const int tid = threadIdx.x;
  const int lane = tid & 31;
  const int wave = tid >> 5;
  const int m0 = blockIdx.y * 64;
  const int n0 = blockIdx.x * 128;
  const int kbeg = blockIdx.z * (INTER / KSPLIT);
  const int kend = kbeg + (INTER / KSPLIT);
  const int wm = (wave & 3) * 16;
  const int wn = (wave >> 2) * 64;
  const int lm = lane & 15;
  const int lg = lane >> 4;

  const int aBase = (wm + lm) * 36 + 2 * lg;
  const int bBase = (wn + lm) * 34 + 2 * lg;
  const unsigned aOff = lds_addr_of(aT);

  v8f acc[4] = {};
  float4 rb[4];

#pragma unroll
  for (int i = 0; i < 4; ++i) {
    const int f = tid * 4 + i * 1024;
    const int r = f >> 7, c = f & 127;
    rb[i] = *(const float4*)(W2 + (size_t)(kbeg + r) * HID + n0 + c);
  }

  for (int kk = kbeg; kk < kend; kk += BK) {
    if (wave == 0) tdm_load_tile_64x32(aOff, A + (size_t)m0 * INTER + kk, INTER);

#pragma unroll
    for (int i = 0; i < 4; ++i) {
      const int f = tid * 4 + i * 1024;
      const int r = f >> 7, c = f & 127;
      bT[(c + 0) * 34 + r] = rb[i].x;
      bT[(c + 1) * 34 + r] = rb[i].y;
      bT[(c + 2) * 34 + r] = rb[i].z;
      bT[(c + 3) * 34 + r] = rb[i].w;
    }
    if (wave == 0) __builtin_amdgcn_s_wait_tensorcnt(0);
    __syncthreads();

    if (kk + BK < kend) {
#pragma unroll
      for (int i = 0; i < 4; ++i) {
        const int f = tid * 4 + i * 1024;
        const int r = f >> 7, c = f & 127;
        rb[i] = *(const float4*)(W2 + (size_t)(kk + BK + r) * HID + n0 + c);
      }
    }

#pragma unroll
    for (int k4 = 0; k4 < BK; k4 += 4) {
      const v2f a = *(const v2f*)&aT[aBase + k4];
#pragma unroll
      for (int nt = 0; nt < 4; ++nt) {
        const v2f b = *(const v2f*)&bT[bBase + nt * 16 * 34 + k4];
        acc[nt] = wmma_f32(a, b, acc[nt]);
      }
    }
    __syncthreads();
  }

#pragma unroll
  for (int nt = 0; nt < 4; ++nt) {
    const int n = n0 + wn + nt * 16 + lm;
#pragma unroll
    for (int v = 0; v < 8; ++v) {
      const int m = m0 + wm + v + 8 * lg;
      const float cf = coef[(size_t)m * NUM_EXPERTS + e];
      atomicAdd(&out[(size_t)m * HID + n], cf * acc[nt][v]);
    }
  }
}

// ---------------------------------------------------------------------------
extern "C" void kernel_launch(void* const* d_in, const int* in_sizes, int n_in,
                              void* d_out, int out_size, void* d_ws, size_t ws_size,
                              hipStream_t stream) {
  const float* X   = (const float*)d_in[0];  // [T, H]
  const float* GW  = (const float*)d_in[1];  // [H, E]
  const float* W1s = (const float*)d_in[2];  // [E, H, I]
  const float* W2s = (const float*)d_in[3];  // [E, I, H]
  const float* W3s = (const float*)d_in[4];  // [E, H, I]
  float* out = (float*)d_out;                // [T, H]

  float* coef = (float*)d_ws;                   // T*E floats
  float* act  = (float*)((char*)d_ws + 65536);  // T*I floats (32 MB)

  moe_router_k<<<NTOK, 256, 0, stream>>>(X, GW, coef);

  const int n4 = out_size / 4;
  moe_zero_k<<<(n4 + 255) / 256, 256, 0, stream>>>((float4*)out, n4);

  for (int e = 0; e < NUM_EXPERTS; ++e) {
    const size_t w13 = (size_t)e * HID * INTER;
    const size_t w2  = (size_t)e * INTER * HID;
    moe_gemm13_k<<<dim3(INTER / 128, NTOK / 64), 256, 0, stream>>>(
        X, W1s + w13, W3s + w13, act);
    moe_gemm2_k<<<dim3(HID / 128, NTOK / 64, 4), 256, 0, stream>>>(
        act, W2s + w2, coef, e, out);
  }
}